// BBoxScoreHead_88527865905482
// MI455X (gfx1250) — compile-verified
//
#include <hip/hip_runtime.h>
#include <math.h>

typedef float v2f __attribute__((ext_vector_type(2)));
typedef float v8f __attribute__((ext_vector_type(8)));

#define B_    64
#define C_    256
#define H_    112
#define W_    112
#define HW_   (H_ * W_)
#define LANG_ 256
#define INF_  (C_ * 2 + LANG_)   // 768
#define HID_  256

struct BoxInfo {
    int rlo, rhi;   // row interval [rlo, rhi)
    int clo, chi;   // col interval [clo, chi)
    float inv_area;
    int pad[3];
};

__device__ __forceinline__ float clamp01(float v) {
    return fminf(fmaxf(v, 0.0f), 1.0f);
}

// ---------------------------------------------------------------------------
// Kernel 1: per-box mask interval + inverse area (exact reference math)
// ---------------------------------------------------------------------------
__global__ void bbox_prep_kernel(const float* __restrict__ boxes_xywh,
                                 BoxInfo* __restrict__ info) {
    int b = threadIdx.x;
    if (b >= B_) return;
    float xc = boxes_xywh[b * 4 + 0];
    float yc = boxes_xywh[b * 4 + 1];
    float w  = boxes_xywh[b * 4 + 2];
    float h  = boxes_xywh[b * 4 + 3];

    float x1 = clamp01(xc - w * 0.5f);
    float y1 = clamp01(yc - h * 0.5f);
    float x2 = clamp01(xc + w * 0.5f);
    float y2 = clamp01(yc + h * 0.5f);

    float xlo = fminf(x1, x2), xhi = fmaxf(x1, x2);
    float ylo = fminf(y1, y2), yhi = fmaxf(y1, y2);
    float ww = fmaxf(xhi - xlo, 1e-6f);
    float hh = fmaxf(yhi - ylo, 1e-6f);
    float cx = (xhi + xlo) * 0.5f;
    float cy = (yhi + ylo) * 0.5f;
    x1 = clamp01(cx - ww * 0.5f);
    x2 = clamp01(cx + ww * 0.5f);
    y1 = clamp01(cy - hh * 0.5f);
    y2 = clamp01(cy + hh * 0.5f);

    float bw = fmaxf(x2 - x1, 1e-4f);
    float bh = fmaxf(y2 - y1, 1e-4f);
    float margin = fminf(fmaxf(sqrtf(bw * bw + bh * bh) * 0.25f, 0.02f), 0.18f);
    float mx1 = clamp01(x1 - margin);
    float my1 = clamp01(y1 - margin);
    float mx2 = clamp01(x2 + margin);
    float my2 = clamp01(y2 + margin);

    // linspace(0,1,H): value i/(H-1). Condition is contiguous -> interval.
    int rlo = H_, rhi = -1;
    for (int i = 0; i < H_; ++i) {
        float yi = (float)i * (1.0f / (float)(H_ - 1));
        if (yi >= my1 && yi <= my2) { if (i < rlo) rlo = i; rhi = i; }
    }
    int clo = W_, chi = -1;
    for (int i = 0; i < W_; ++i) {
        float xi = (float)i * (1.0f / (float)(W_ - 1));
        if (xi >= mx1 && xi <= mx2) { if (i < clo) clo = i; chi = i; }
    }
    int rows = (rhi >= rlo) ? (rhi - rlo + 1) : 0;
    int cols = (chi >= clo) ? (chi - clo + 1) : 0;
    float area = fmaxf((float)(rows * cols), 1.0f);

    BoxInfo bi;
    bi.rlo = (rows > 0) ? rlo : 0;
    bi.rhi = (rows > 0) ? rhi + 1 : 0;
    bi.clo = (cols > 0) ? clo : 0;
    bi.chi = (cols > 0) ? chi + 1 : 0;
    bi.inv_area = 1.0f / area;
    bi.pad[0] = bi.pad[1] = bi.pad[2] = 0;
    info[b] = bi;
}

// ---------------------------------------------------------------------------
// Kernel 2: copy lang_vec into combined[:, 512:768]
// ---------------------------------------------------------------------------
__global__ void lang_copy_kernel(const float* __restrict__ lang,
                                 float* __restrict__ combined) {
    int b = blockIdx.x;
    int c = threadIdx.x;
    combined[b * INF_ + 2 * C_ + c] = lang[b * LANG_ + c];
}

// ---------------------------------------------------------------------------
// Kernel 3: streaming masked + global pooling. One block per (b,c) plane.
// 822 MB streamed once; float4 (b128) loads; dual accumulate; LDS reduce.
// ---------------------------------------------------------------------------
__global__ void pool_kernel(const float* __restrict__ feat,
                            const BoxInfo* __restrict__ info,
                            float* __restrict__ combined) {
    int plane = blockIdx.x;            // b*C + c
    int b = plane >> 8;                // C_ == 256
    int c = plane & (C_ - 1);
    int tid = threadIdx.x;

    const float4* p = (const float4*)(feat + (size_t)plane * HW_);
    BoxInfo bi = info[b];

    float tot = 0.0f, msk = 0.0f;
    // W_ == 112 is a multiple of 4 -> each float4 lies within one row.
    for (int i = tid; i < HW_ / 4; i += blockDim.x) {
        float4 v = p[i];
        float s4 = (v.x + v.y) + (v.z + v.w);
        tot += s4;
        int idx = i * 4;
        int hh = idx / W_;
        int wc = idx - hh * W_;
        if (hh >= bi.rlo && hh < bi.rhi) {
            if (wc >= bi.clo && wc + 4 <= bi.chi) {
                msk += s4;                       // fully inside
            } else {
                if (wc + 0 >= bi.clo && wc + 0 < bi.chi) msk += v.x;
                if (wc + 1 >= bi.clo && wc + 1 < bi.chi) msk += v.y;
                if (wc + 2 >= bi.clo && wc + 2 < bi.chi) msk += v.z;
                if (wc + 3 >= bi.clo && wc + 3 < bi.chi) msk += v.w;
            }
        }
    }

    __shared__ float st[256];
    __shared__ float sm[256];
    st[tid] = tot;
    sm[tid] = msk;
    __syncthreads();
    for (int s = 128; s > 0; s >>= 1) {
        if (tid < s) { st[tid] += st[tid + s]; sm[tid] += sm[tid + s]; }
        __syncthreads();
    }
    if (tid == 0) {
        combined[b * INF_ + c]      = sm[0] * bi.inv_area;          // pooled
        combined[b * INF_ + C_ + c] = st[0] * (1.0f / (float)HW_);  // global avg
    }
}

// ---------------------------------------------------------------------------
// Kernel 4: Y = act(X @ W^T + bias) via v_wmma_f32_16x16x4_f32.
// One wave (32 threads) per 16x16 output tile; EXEC all ones as required.
// A layout (16x4 f32): element (m,k) -> lane = m + 16*(k>=2), vgpr = k&1.
// B layout (4x16 f32): element (k,n) -> lane = n + 16*(k>=2), vgpr = k&1.
// C/D layout: vgpr r, lane l -> (m = r + 8*(l>=16), n = l&15).
// ---------------------------------------------------------------------------
__global__ void mlp_wmma_kernel(const float* __restrict__ X,   // M x K row-major
                                const float* __restrict__ Wt,  // N x K row-major (use W^T)
                                const float* __restrict__ bias,
                                float* __restrict__ Y,         // M x N
                                int M, int N, int K, int do_relu) {
    int lane = threadIdx.x;            // 0..31, one wave per block
    int tilesM = M >> 4;
    int m0 = (blockIdx.x % tilesM) << 4;
    int n0 = (blockIdx.x / tilesM) << 4;
    int mm = lane & 15;
    int hh = lane >> 4;                // K half-select

    const float* arow = X  + (size_t)(m0 + mm) * K;
    const float* brow = Wt + (size_t)(n0 + mm) * K;

    v8f acc = {};
    for (int k0 = 0; k0 < K; k0 += 4) {
        int kk = k0 + 2 * hh;
        v2f a = { arow[kk], arow[kk + 1] };
        v2f b = { brow[kk], brow[kk + 1] };
        acc = __builtin_amdgcn_wmma_f32_16x16x4_f32(
            /*neg_a=*/false, a, /*neg_b=*/false, b,
            /*c_mod=*/(short)0, acc, /*reuse_a=*/false, /*reuse_b=*/false);
    }

    float bv = bias[n0 + mm];
#pragma unroll
    for (int r = 0; r < 8; ++r) {
        float v = acc[r] + bv;
        if (do_relu) v = fmaxf(v, 0.0f);
        Y[(size_t)(m0 + r + 8 * hh) * N + (n0 + mm)] = v;
    }
}

// ---------------------------------------------------------------------------
// Kernel 5: out[b] = sigmoid(x2[b,:] . w3 + b3)
// ---------------------------------------------------------------------------
__global__ void head_kernel(const float* __restrict__ x2,
                            const float* __restrict__ w3,
                            const float* __restrict__ b3,
                            float* __restrict__ out) {
    int b = threadIdx.x;
    if (b >= B_) return;
    float s = b3[0];
    for (int k = 0; k < HID_; ++k) s += x2[b * HID_ + k] * w3[k];
    out[b] = 1.0f / (1.0f + expf(-s));
}

// ---------------------------------------------------------------------------
extern "C" void kernel_launch(void* const* d_in, const int* in_sizes, int n_in,
                              void* d_out, int out_size, void* d_ws, size_t ws_size,
                              hipStream_t stream) {
    const float* feat       = (const float*)d_in[0];
    const float* lang_vec   = (const float*)d_in[1];
    const float* boxes_xywh = (const float*)d_in[2];
    const float* w1         = (const float*)d_in[3];
    const float* b1         = (const float*)d_in[4];
    const float* w2         = (const float*)d_in[5];
    const float* b2         = (const float*)d_in[6];
    const float* w3         = (const float*)d_in[7];
    const float* b3         = (const float*)d_in[8];
    float* out = (float*)d_out;

    // Workspace layout
    BoxInfo* info   = (BoxInfo*)d_ws;                               // 64 * 32 B
    float* combined = (float*)((char*)d_ws + B_ * sizeof(BoxInfo)); // 64 x 768
    float* x1       = combined + (size_t)B_ * INF_;                 // 64 x 256
    float* x2       = x1 + (size_t)B_ * HID_;                       // 64 x 256

    // 1. Box -> mask interval
    bbox_prep_kernel<<<1, 64, 0, stream>>>(boxes_xywh, info);

    // 2. lang_vec into combined[:, 512:768]
    lang_copy_kernel<<<B_, LANG_, 0, stream>>>(lang_vec, combined);

    // 3. Streaming pooling: one block per (b,c) plane
    pool_kernel<<<B_ * C_, 256, 0, stream>>>(feat, info, combined);

    // 4. MLP layer 1: (64x768) @ (768x256) + b1, relu  -> x1
    {
        int M = B_, N = HID_, K = INF_;
        int grid = (M / 16) * (N / 16);   // 4 * 16 = 64 waves
        mlp_wmma_kernel<<<grid, 32, 0, stream>>>(combined, w1, b1, x1, M, N, K, 1);
    }
    // 5. MLP layer 2: (64x256) @ (256x256) + b2, relu -> x2
    {
        int M = B_, N = HID_, K = HID_;
        int grid = (M / 16) * (N / 16);
        mlp_wmma_kernel<<<grid, 32, 0, stream>>>(x1, w2, b2, x2, M, N, K, 1);
    }
    // 6. Head: sigmoid(x2 @ w3^T + b3) -> out (64 floats)
    head_kernel<<<1, 64, 0, stream>>>(x2, w3, b3, out);
}